// ClassCapsule_26989574488808
// MI455X (gfx1250) — compile-verified
//
#include <hip/hip_runtime.h>
#include <hip/hip_bf16.h>

// Problem constants (from reference)
#define BB    32
#define CC    32
#define DIN   16
#define KCAP  10
#define DOUT  16
#define HH    14
#define WW    14
#define HW    196          // 14*14
#define NO    160          // KCAP*DOUT
#define MPAD  208          // HW padded to multiple of 16 (votes M-stride)
#define MTILES 13          // 208/16
#define NTILES 10          // 160/16
#define PI_CONST 3.1416f

typedef float v2f __attribute__((ext_vector_type(2)));
typedef float v8f __attribute__((ext_vector_type(8)));
typedef unsigned int u32x4 __attribute__((ext_vector_type(4)));
typedef int i32x4 __attribute__((ext_vector_type(4)));
typedef int i32x8 __attribute__((ext_vector_type(8)));

#if defined(__has_builtin)
#if __has_builtin(__builtin_amdgcn_tensor_load_to_lds) && __has_builtin(__builtin_amdgcn_s_wait_tensorcnt)
#define USE_TDM 1
#endif
#endif
#ifndef USE_TDM
#define USE_TDM 0
#endif

#if USE_TDM
// Issue a 1-D TDM copy of `ndw` dwords from global `gaddr` into LDS byte offset
// `lds_off`. Optional LDS padding: insert (pad_amount+1) dwords every
// 2^(pad_interval+1) dwords (pad_interval encoding per D# group1).
__device__ __forceinline__ void tdm_load_1d(unsigned lds_off, const void* gaddr,
                                            unsigned ndw, int pad_en,
                                            int pad_interval, int pad_amount)
{
    unsigned long long ga = (unsigned long long)gaddr;
    u32x4 g0;
    g0[0] = 1u;                                        // count=1, user descriptor
    g0[1] = lds_off;                                   // lds_addr (bytes)
    g0[2] = (unsigned)(ga & 0xFFFFFFFFu);              // global_addr[31:0]
    g0[3] = (unsigned)((ga >> 32) & 0x1FFFFFFu)        // global_addr[56:32]
          | (2u << 30);                                // type = 2 ("image")
    i32x8 g1;
    g1[0] = (int)((2u << 16)                           // data_size = 4B
          | (pad_en ? (1u << 20) : 0u)                 // pad_enable
          | (((unsigned)pad_interval & 7u) << 22)
          | (((unsigned)pad_amount & 127u) << 25));
    g1[1] = (int)((ndw & 0xFFFFu) << 16);              // tensor_dim0[15:0]
    g1[2] = (int)(((ndw >> 16) & 0xFFFFu)              // tensor_dim0[31:16]
          | (1u << 16));                               // tensor_dim1 = 1
    g1[3] = (int)((ndw & 0xFFFFu) << 16);              // tile_dim0 = ndw
    g1[4] = 0;                                         // tile_dim1/2 = 0 (1-D)
    g1[5] = (int)ndw;                                  // tensor_dim0_stride
    g1[6] = 0;
    g1[7] = 0;
#if __clang_major__ >= 23
    __builtin_amdgcn_tensor_load_to_lds(g0, g1, (i32x4)0, (i32x4)0, (i32x8)0, 0);
#else
    __builtin_amdgcn_tensor_load_to_lds(g0, g1, (i32x4)0, (i32x4)0, 0);
#endif
}
#endif

// ---------------------------------------------------------------------------
// Kernel 1: votes[b,c,o,m] = sum_i vec[b,c,i,m] * weight[c,o,i] + bias[c,o]
// One block per (b,c); operands staged to LDS via TDM; f32 WMMA 16x16x4 x4.
// votes stored with padded M-stride (MPAD) so all stores are 2x b128,
// unconditional (padded rows are write-only garbage, never read back).
// ---------------------------------------------------------------------------
__global__ __launch_bounds__(256) void votes_kernel(
    const float* __restrict__ x,
    const float* __restrict__ weight,
    const float* __restrict__ bias,
    float* __restrict__ votes)
{
    __shared__ float sVec[DIN * HW];    // raw contiguous vec tile (TDM target)
    __shared__ float sW[NO][17];        // weight, row-padded 16->17 via TDM pad
    __shared__ float sBias[NO];

    const int blk = blockIdx.x;         // b*CC + c
    const int b   = blk / CC;
    const int c   = blk % CC;
    const int tid = threadIdx.x;

    const float* vecg = x + ((size_t)b * (CC * 17) + (size_t)c * 17 + 1) * HW;
    const float* wg   = weight + (size_t)c * NO * DIN;
    const float* bg   = bias   + (size_t)c * NO;

#if USE_TDM
    if (tid < 32) {   // one wave issues the three DMA descriptors
        tdm_load_1d((unsigned)(unsigned long long)(void*)&sVec[0],
                    vecg, DIN * HW, 0, 0, 0);
        // weight: pad 1 dword after every 16 dwords -> lands as [160][17]
        tdm_load_1d((unsigned)(unsigned long long)(void*)&sW[0][0],
                    wg, NO * DIN, 1, /*interval 16dw*/3, /*1 dword*/0);
        tdm_load_1d((unsigned)(unsigned long long)(void*)&sBias[0],
                    bg, NO, 0, 0, 0);
        __builtin_amdgcn_s_wait_tensorcnt(0);
    }
    __syncthreads();
    // coord fix-up: vec[0] += h/10, vec[1] += w/10
    for (int idx = tid; idx < 2 * HW; idx += 256) {
        int i = idx / HW, m = idx % HW;
        sVec[i * HW + m] += (i == 0) ? (float)(m / WW) * 0.1f
                                     : (float)(m % WW) * 0.1f;
    }
#else
    for (int idx = tid; idx < NO * DIN; idx += 256)
        sW[idx / DIN][idx % DIN] = wg[idx];
    if (tid < NO) sBias[tid] = bg[tid];
    for (int idx = tid; idx < DIN * HW; idx += 256) {
        int i = idx / HW, m = idx % HW;
        float v = vecg[idx];
        if (i == 0)      v += (float)(m / WW) * 0.1f;
        else if (i == 1) v += (float)(m % WW) * 0.1f;
        sVec[idx] = v;
    }
#endif
    __syncthreads();

    const int wave  = tid >> 5;
    const int lane  = tid & 31;
    const int lhalf = lane >> 4;   // 0: K pair {k,k+1}, 1: K pair {k+2,k+3}
    const int l15   = lane & 15;

    float* out = votes + (size_t)blk * NO * MPAD;

    for (int t = wave; t < MTILES * NTILES; t += 8) {
        const int mt = t / NTILES, nt = t % NTILES;
        const int m0 = mt * 16,    n0 = nt * 16;
        // clamp M index for the padded tail tile; resulting garbage rows land
        // in the write-only padded region of votes
        const int mi = min(m0 + l15, HW - 1);

        v8f acc = {};
#pragma unroll
        for (int kk = 0; kk < DIN; kk += 4) {
            // A (16x4 f32): lanes 0-15 hold K=kk,kk+1 of row M; lanes 16-31 K=kk+2,kk+3
            v2f afrag, bfrag;
            afrag.x = sVec[(kk + 2 * lhalf    ) * HW + mi];
            afrag.y = sVec[(kk + 2 * lhalf + 1) * HW + mi];
            // B (4x16 f32): row-striped across lanes, same K split
            bfrag.x = sW[n0 + l15][kk + 2 * lhalf    ];
            bfrag.y = sW[n0 + l15][kk + 2 * lhalf + 1];
            acc = __builtin_amdgcn_wmma_f32_16x16x4_f32(
                false, afrag, false, bfrag, (short)0, acc, false, false);
        }

        // D layout: VGPR r -> M = m0 + lhalf*8 + r, N = n0 + l15 (fixed per lane)
        const int   n     = n0 + l15;
        const int   mbase = m0 + lhalf * 8;
        const float badd  = sBias[n];
        float* op = out + (size_t)n * MPAD + mbase;
        float4 lo = make_float4(acc[0] + badd, acc[1] + badd, acc[2] + badd, acc[3] + badd);
        float4 hi = make_float4(acc[4] + badd, acc[5] + badd, acc[6] + badd, acc[7] + badd);
        *(float4*)(op)     = lo;
        *(float4*)(op + 4) = hi;
    }
}

// ---------------------------------------------------------------------------
// Kernel 2: per (b,k) moment reduction over (c,hw):
//   S0 = sum r_hat, S1[d] = sum r_hat*v, S2[d] = sum r_hat*v^2
//   u = S1/S0 ; sigma = S2/S0 - u^2 ; cost ; a_hat
// Writes stats[b*K+k] = {u[16], 0.5/sigma[16], logZ, log(a_hat)} and a_hat.
// ---------------------------------------------------------------------------
__global__ __launch_bounds__(256) void em_reduce_kernel(
    const float* __restrict__ x,        // act at channel c*17
    const float* __restrict__ votes,
    const float* __restrict__ R,
    const float* __restrict__ beta_v,
    const float* __restrict__ beta_a,
    const int*   __restrict__ lamda,
    float* __restrict__ stats,          // [B*K][34]
    float* __restrict__ ahat_out,       // [B*K]
    int uniformR)
{
    const int blk = blockIdx.x;         // b*KCAP + k
    const int b   = blk / KCAP;
    const int k   = blk % KCAP;
    const int tid = threadIdx.x;

    float s0 = 0.0f, s1[DOUT], s2[DOUT];
#pragma unroll
    for (int d = 0; d < DOUT; ++d) { s1[d] = 0.0f; s2[d] = 0.0f; }

    const float* actb   = x     + (size_t)b * (CC * 17) * HW;
    const float* votesb = votes + (size_t)b * CC * NO * MPAD;
    const float* Rb     = R     + (size_t)b * CC * KCAP * HW;

    for (int idx = tid; idx < CC * HW; idx += 256) {
        const int c = idx / HW, hw = idx % HW;
        const float act = actb[(size_t)c * 17 * HW + hw];
        const float r   = uniformR ? (1.0f / (float)KCAP)
                                   : Rb[((size_t)c * KCAP + k) * HW + hw];
        const float rh  = r * act;
        s0 += rh;
        const float* vp = votesb + ((size_t)c * NO + (size_t)k * DOUT) * MPAD + hw;
        __builtin_prefetch(vp + 256, 0, 0);
#pragma unroll
        for (int d = 0; d < DOUT; ++d) {
            const float v = vp[(size_t)d * MPAD];
            s1[d] += rh * v;
            s2[d] += rh * v * v;
        }
    }

    // wave32 shuffle reduction
#pragma unroll
    for (int off = 16; off >= 1; off >>= 1) {
        s0 += __shfl_down(s0, off, 32);
#pragma unroll
        for (int d = 0; d < DOUT; ++d) {
            s1[d] += __shfl_down(s1[d], off, 32);
            s2[d] += __shfl_down(s2[d], off, 32);
        }
    }

    __shared__ float red[8][33];
    __shared__ float fin[33];
    const int wave = tid >> 5, lane = tid & 31;
    if (lane == 0) {
        red[wave][0] = s0;
#pragma unroll
        for (int d = 0; d < DOUT; ++d) {
            red[wave][1 + d]        = s1[d];
            red[wave][1 + DOUT + d] = s2[d];
        }
    }
    __syncthreads();
    if (tid < 33) {
        float acc = 0.0f;
        for (int w = 0; w < 8; ++w) acc += red[w][tid];
        fin[tid] = acc;
    }
    __syncthreads();

    if (tid == 0) {
        const float S0  = fin[0];
        const float inv = 1.0f / S0;
        float cost = 0.0f, logZ = 0.0f;
        float* st = stats + (size_t)blk * 34;
#pragma unroll
        for (int d = 0; d < DOUT; ++d) {
            const float u   = fin[1 + d] * inv;
            float sig       = fin[1 + DOUT + d] * inv - u * u;
            sig             = fmaxf(sig, 1e-12f);
            st[d]           = u;
            st[DOUT + d]    = 0.5f / sig;
            const float ls  = __logf(sig);
            cost += (beta_v[k] + ls) * S0;
            logZ += 0.5f * (__logf(2.0f * PI_CONST) + ls);
        }
        st[32] = logZ;
        const float lam = (float)lamda[0];
        const float ah  = 1.0f / (1.0f + __expf(-lam * (beta_a[k] - cost)));
        ahat_out[blk] = ah;
        st[33] = __logf(fmaxf(ah, 1e-30f));
    }
}

// ---------------------------------------------------------------------------
// Kernel 3: R[b,c,k,hw] = softmax_k( log a_hat - sum_d diff^2 * inv2sig - logZ )
// Grid: (ceil(C*HW/256), B). Per-b stats cached in LDS.
// ---------------------------------------------------------------------------
__global__ __launch_bounds__(256) void r_update_kernel(
    const float* __restrict__ votes,
    const float* __restrict__ stats,    // [B*K][34]
    float* __restrict__ R)
{
    const int b   = blockIdx.y;
    const int tid = threadIdx.x;

    __shared__ float sst[KCAP * 34];
    for (int i = tid; i < KCAP * 34; i += 256)
        sst[i] = stats[(size_t)b * KCAP * 34 + i];
    __syncthreads();

    const int idx = blockIdx.x * 256 + tid;
    if (idx >= CC * HW) return;
    const int c  = idx / HW;
    const int hw = idx % HW;

    const float* vb = votes + ((size_t)b * CC + c) * NO * MPAD + hw;

    float logits[KCAP];
    float mx = -1e30f;
#pragma unroll
    for (int k = 0; k < KCAP; ++k) {
        const float* st = &sst[k * 34];
        float s = 0.0f;
#pragma unroll
        for (int d = 0; d < DOUT; ++d) {
            const float diff = vb[(size_t)(k * DOUT + d) * MPAD] - st[d];
            s += diff * diff * st[DOUT + d];
        }
        const float lg = st[33] - s - st[32];   // log a_hat + log p_c
        logits[k] = lg;
        mx = fmaxf(mx, lg);
    }
    float e[KCAP], sum = 0.0f;
#pragma unroll
    for (int k = 0; k < KCAP; ++k) { e[k] = __expf(logits[k] - mx); sum += e[k]; }
    const float invs = 1.0f / sum;

    float* Rp = R + ((size_t)b * CC + c) * KCAP * HW + hw;
#pragma unroll
    for (int k = 0; k < KCAP; ++k) Rp[(size_t)k * HW] = e[k] * invs;
}

// ---------------------------------------------------------------------------
extern "C" void kernel_launch(void* const* d_in, const int* in_sizes, int n_in,
                              void* d_out, int out_size, void* d_ws, size_t ws_size,
                              hipStream_t stream)
{
    (void)in_sizes; (void)n_in; (void)out_size; (void)ws_size;
    const float* x      = (const float*)d_in[0];
    const float* weight = (const float*)d_in[1];
    const float* bias   = (const float*)d_in[2];
    const float* beta_v = (const float*)d_in[3];
    const float* beta_a = (const float*)d_in[4];
    const int*   lamda  = (const int*)d_in[5];
    float* out = (float*)d_out;

    // Workspace layout
    float* votes = (float*)d_ws;                                  // B*C*160*208 (padded M)
    size_t votesN = (size_t)BB * CC * NO * MPAD;
    float* R     = votes + votesN;                                // B*C*K*196
    size_t RN     = (size_t)BB * CC * KCAP * HW;
    float* stats = R + RN;                                        // B*K*34
    float* ahat  = stats + (size_t)BB * KCAP * 34;                // B*K

    votes_kernel<<<BB * CC, 256, 0, stream>>>(x, weight, bias, votes);

    const int rblocks = (CC * HW + 255) / 256;   // 25
    for (int it = 0; it < 3; ++it) {
        float* adest = (it == 2) ? out : ahat;
        em_reduce_kernel<<<BB * KCAP, 256, 0, stream>>>(
            x, votes, R, beta_v, beta_a, lamda, stats, adest, (it == 0) ? 1 : 0);
        if (it < 2)
            r_update_kernel<<<dim3(rblocks, BB), 256, 0, stream>>>(votes, stats, R);
    }
}